// FARSRadialFrequencyFinal_23613730194171
// MI455X (gfx1250) — compile-verified
//
#include <hip/hip_runtime.h>
#include <math.h>

#define D_LAT 448
#define G_GRAPHS 4096
#define SID_K 1794
#define KP 1856          // K padded to multiple of 64 -> constant chunk, full unroll
#define LN_EPS 1e-5f

typedef float v2f __attribute__((ext_vector_type(2)));
typedef float v8f __attribute__((ext_vector_type(8)));

__device__ __forceinline__ float wave_reduce_add(float v) {
  #pragma unroll
  for (int o = 16; o > 0; o >>= 1) v += __shfl_down(v, o, 32);
  return v;
}

// -------------------------------------------------------------------------
// Kernel A: per-graph shell-0 pooling (sum/max/count) + feature assembly +
// LayerNorm1, writing the padded GEMM1 activation row [G, KP].
// batch[] is sorted -> one block per graph, binary search for node range.
// Each (slot, dim) accumulator is owned by exactly one thread: no atomics.
// -------------------------------------------------------------------------
__global__ __launch_bounds__(448)
void pool_ln_kernel(const float* __restrict__ x,
                    const int*   __restrict__ batch,
                    const int*   __restrict__ shell,
                    const int*   __restrict__ side,
                    const float* __restrict__ g1,
                    const float* __restrict__ b1,
                    float*       __restrict__ Apad,
                    int Nn) {
  __shared__ float s_sum[2 * D_LAT];
  __shared__ float s_max[2 * D_LAT];
  __shared__ float s_part[2][16];
  __shared__ float s_stats[2];
  __shared__ int   s_cnt[2];
  __shared__ int   s_range[2];

  const int g = blockIdx.x;
  const int t = threadIdx.x;

  if (t == 0) {
    int lo = 0, hi = Nn;
    while (lo < hi) { int m = (lo + hi) >> 1; if (batch[m] < g) lo = m + 1; else hi = m; }
    s_range[0] = lo;
    int lo2 = lo, hi2 = Nn;
    while (lo2 < hi2) { int m = (lo2 + hi2) >> 1; if (batch[m] < g + 1) lo2 = m + 1; else hi2 = m; }
    s_range[1] = lo2;
    s_cnt[0] = 0; s_cnt[1] = 0;
  }
  s_sum[t] = 0.f;            s_sum[D_LAT + t] = 0.f;
  s_max[t] = -INFINITY;      s_max[D_LAT + t] = -INFINITY;
  __syncthreads();

  const int lo = s_range[0], hi = s_range[1];
  for (int n = lo; n < hi; ++n) {
    int sh = shell[n];                      // uniform -> scalar load
    if (sh == 0) {                          // only shell 0 reaches the output
      int sd = side[n];
      float v = x[(size_t)n * D_LAT + t];
      int o = sd * D_LAT + t;
      s_sum[o] += v;
      s_max[o] = fmaxf(s_max[o], v);
      if (t == 0) s_cnt[sd]++;
    }
  }
  __syncthreads();

  const float c0 = (float)s_cnt[0], c1 = (float)s_cnt[1];
  float mean0 = s_sum[t]         / fmaxf(c0, 1.f);
  float max0  = (c0 > 0.f) ? s_max[t]         : 0.f;
  float mean1 = s_sum[D_LAT + t] / fmaxf(c1, 1.f);
  float max1  = (c1 > 0.f) ? s_max[D_LAT + t] : 0.f;
  float lc = (t == 0) ? log1pf(c0) : ((t == 1) ? log1pf(c1) : 0.f);

  float ls  = mean0 + max0 + mean1 + max1 + lc;
  float ls2 = mean0*mean0 + max0*max0 + mean1*mean1 + max1*max1 + lc*lc;
  ls  = wave_reduce_add(ls);
  ls2 = wave_reduce_add(ls2);
  const int wid = t >> 5, lane = t & 31;
  if (lane == 0) { s_part[0][wid] = ls; s_part[1][wid] = ls2; }
  __syncthreads();
  if (t == 0) {
    float S = 0.f, S2 = 0.f;
    for (int i = 0; i < 14; ++i) { S += s_part[0][i]; S2 += s_part[1][i]; }
    float mu  = S / (float)SID_K;
    float var = S2 / (float)SID_K - mu * mu;
    s_stats[0] = mu;
    s_stats[1] = rsqrtf(var + LN_EPS);
  }
  __syncthreads();
  const float mu = s_stats[0], rstd = s_stats[1];

  float* row = Apad + (size_t)g * KP;
  int o;
  o = t;             row[o] = (mean0 - mu) * rstd * g1[o] + b1[o];
  o = D_LAT + t;     row[o] = (max0  - mu) * rstd * g1[o] + b1[o];
  o = 2*D_LAT + t;   row[o] = (mean1 - mu) * rstd * g1[o] + b1[o];
  o = 3*D_LAT + t;   row[o] = (max1  - mu) * rstd * g1[o] + b1[o];
  if (t < 2) {
    o = 4*D_LAT + t; row[o] = (lc - mu) * rstd * g1[o] + b1[o];
  }
  if (t < KP - SID_K) row[SID_K + t] = 0.f;   // zero pad K 1794..1855
}

// Zero-pad W_proj [1794,448] -> Wp [1856,448]
__global__ void pad_w_kernel(const float* __restrict__ W, float* __restrict__ Wp) {
  int i = blockIdx.x * blockDim.x + threadIdx.x;
  const int total = KP * D_LAT;
  if (i < total) {
    int k = i / D_LAT;
    Wp[i] = (k < SID_K) ? W[i] : 0.f;
  }
}

// -------------------------------------------------------------------------
// fp32 WMMA GEMM: C[M,N] = act(A[M,K] @ B[K,N] + bias). K % 64 == 0,
// M % 32 == 0, N % 64 == 0, all compile-time (N, K) for immediate offsets.
// 128 threads (4 waves); block tile 32(M) x 64(N); each wave owns two 16x16
// C tiles stacked in M, sharing each B fragment. Double-buffered LDS with
// register prefetch: global loads for chunk k+1 issue before the 32 WMMAs of
// chunk k, so DRAM latency hides behind matrix work; one barrier per chunk.
// V_WMMA_F32_16X16X4_F32: exact fp32 math matching the reference.
// -------------------------------------------------------------------------
template <bool RELU, int N, int K>
__global__ __launch_bounds__(128)
void gemm_wmma(const float* __restrict__ A, const float* __restrict__ B,
               const float* __restrict__ bias, float* __restrict__ C) {
  __shared__ float As[2][32 * 68];   // +4 pad breaks stride-64 bank pattern
  __shared__ float Bs[2][64 * 68];

  const int t = threadIdx.x;
  const int lane = t & 31, w = t >> 5;
  const int m0 = blockIdx.y * 32;
  const int n0 = blockIdx.x * 64;

  // per-thread staging coordinates
  const int rA = t >> 2, cA = (t & 3) * 16;     // A: row rA, 16 cols from cA
  const int rB = t >> 4, cB = (t & 15) * 4;     // B: rows rB+8i, 4 cols from cB
  const float* pA = A + (size_t)(m0 + rA) * K + cA;
  const float* pB = B + (size_t)rB * N + n0 + cB;

  float4 aReg[4];
  float4 bReg[8];

  // ---- preload chunk 0 into registers, stage to buffer 0 ----
  #pragma unroll
  for (int j = 0; j < 4; ++j) aReg[j] = *(const float4*)(pA + j * 4);
  #pragma unroll
  for (int i = 0; i < 8; ++i) bReg[i] = *(const float4*)(pB + i * 8 * N);
  pA += 64;
  pB += (size_t)64 * N;
  {
    float* dA = &As[0][rA * 68 + cA];
    #pragma unroll
    for (int j = 0; j < 4; ++j) *(float4*)(dA + j * 4) = aReg[j];
    #pragma unroll
    for (int i = 0; i < 8; ++i) *(float4*)&Bs[0][(rB + i * 8) * 68 + cB] = bReg[i];
  }
  __syncthreads();

  v8f acc0 = {};
  v8f acc1 = {};

  const int mrow = lane & 15;
  const int ks   = (lane >> 4) * 2;
  const int nn   = (lane & 15) + w * 16;

  int cur = 0;
  for (int k0 = 0; k0 < K; k0 += 64) {
    const bool more = (k0 + 64) < K;
    // ---- issue global prefetch of next chunk (no wait here) ----
    if (more) {
      #pragma unroll
      for (int j = 0; j < 4; ++j) aReg[j] = *(const float4*)(pA + j * 4);
      #pragma unroll
      for (int i = 0; i < 8; ++i) bReg[i] = *(const float4*)(pB + i * 8 * N);
      pA += 64;
      pB += (size_t)64 * N;
    }
    // ---- compute on current buffer: 16 k-steps x 2 M-tiles = 32 WMMA ----
    {
      const float* as = As[cur];
      const float* bs = Bs[cur];
      #pragma unroll
      for (int kk = 0; kk < 64; kk += 4) {
        v2f a0, a1, b;
        a0.x = as[mrow * 68 + kk + ks];
        a0.y = as[mrow * 68 + kk + ks + 1];
        a1.x = as[(16 + mrow) * 68 + kk + ks];
        a1.y = as[(16 + mrow) * 68 + kk + ks + 1];
        b.x  = bs[(kk + ks) * 68 + nn];
        b.y  = bs[(kk + ks + 1) * 68 + nn];
        acc0 = __builtin_amdgcn_wmma_f32_16x16x4_f32(
            false, a0, false, b, (short)0, acc0, false, false);
        acc1 = __builtin_amdgcn_wmma_f32_16x16x4_f32(
            false, a1, false, b, (short)0, acc1, false, false);
      }
    }
    // ---- stage prefetched registers into the other buffer ----
    if (more) {
      const int nxt = cur ^ 1;
      float* dA = &As[nxt][rA * 68 + cA];
      #pragma unroll
      for (int j = 0; j < 4; ++j) *(float4*)(dA + j * 4) = aReg[j];
      #pragma unroll
      for (int i = 0; i < 8; ++i) *(float4*)&Bs[nxt][(rB + i * 8) * 68 + cB] = bReg[i];
      __syncthreads();
      cur = nxt;
    }
  }

  // Epilogue: bias + optional ReLU, scatter per C/D VGPR layout
  const int ncol  = n0 + w * 16 + (lane & 15);
  const int mbase = m0 + ((lane < 16) ? 0 : 8);
  const float bv = bias ? bias[ncol] : 0.f;
  #pragma unroll
  for (int r = 0; r < 8; ++r) {
    float v = acc0[r] + bv;
    if (RELU) v = fmaxf(v, 0.f);
    C[(size_t)(mbase + r) * N + ncol] = v;
  }
  #pragma unroll
  for (int r = 0; r < 8; ++r) {
    float v = acc1[r] + bv;
    if (RELU) v = fmaxf(v, 0.f);
    C[(size_t)(mbase + 16 + r) * N + ncol] = v;
  }
}

// LayerNorm2 over D=448 per graph row
__global__ __launch_bounds__(128)
void ln2_kernel(const float* __restrict__ H, const float* __restrict__ g2,
                const float* __restrict__ b2, float* __restrict__ O) {
  __shared__ float s_part[2][4];
  __shared__ float s_stats[2];
  const int g = blockIdx.x, t = threadIdx.x;
  const float* row = H + (size_t)g * D_LAT;
  float v[4];
  float s = 0.f, s2 = 0.f;
  int cnt = 0;
  for (int i = t; i < D_LAT; i += 128) { float xv = row[i]; v[cnt++] = xv; s += xv; s2 += xv * xv; }
  s = wave_reduce_add(s); s2 = wave_reduce_add(s2);
  const int lane = t & 31, wid = t >> 5;
  if (lane == 0) { s_part[0][wid] = s; s_part[1][wid] = s2; }
  __syncthreads();
  if (t == 0) {
    float S  = s_part[0][0] + s_part[0][1] + s_part[0][2] + s_part[0][3];
    float S2 = s_part[1][0] + s_part[1][1] + s_part[1][2] + s_part[1][3];
    float mu = S / 448.f;
    float var = S2 / 448.f - mu * mu;
    s_stats[0] = mu; s_stats[1] = rsqrtf(var + LN_EPS);
  }
  __syncthreads();
  const float mu = s_stats[0], rstd = s_stats[1];
  float* orow = O + (size_t)g * D_LAT;
  cnt = 0;
  for (int i = t; i < D_LAT; i += 128) orow[i] = (v[cnt++] - mu) * rstd * g2[i] + b2[i];
}

// out[g] = sum_t H2[g,t] * W2[t] + b2
__global__ __launch_bounds__(128)
void readout_kernel(const float* __restrict__ H2, const float* __restrict__ W2,
                    const float* __restrict__ b2, float* __restrict__ out) {
  __shared__ float s_part[4];
  const int g = blockIdx.x, t = threadIdx.x;
  float v = H2[(size_t)g * 128 + t] * W2[t];
  v = wave_reduce_add(v);
  const int lane = t & 31, wid = t >> 5;
  if (lane == 0) s_part[wid] = v;
  __syncthreads();
  if (t == 0) out[g] = s_part[0] + s_part[1] + s_part[2] + s_part[3] + b2[0];
}

extern "C" void kernel_launch(void* const* d_in, const int* in_sizes, int n_in,
                              void* d_out, int out_size, void* d_ws, size_t ws_size,
                              hipStream_t stream) {
  (void)n_in; (void)out_size; (void)ws_size;
  const float* x      = (const float*)d_in[0];
  const int*   batch  = (const int*)d_in[1];
  const int*   shell  = (const int*)d_in[2];
  const int*   side   = (const int*)d_in[3];
  const float* ln1_g  = (const float*)d_in[4];
  const float* ln1_b  = (const float*)d_in[5];
  const float* W_proj = (const float*)d_in[6];
  const float* b_proj = (const float*)d_in[7];
  const float* ln2_g  = (const float*)d_in[8];
  const float* ln2_b  = (const float*)d_in[9];
  const float* W1     = (const float*)d_in[10];
  const float* b1     = (const float*)d_in[11];
  const float* W2     = (const float*)d_in[12];
  const float* b2     = (const float*)d_in[13];
  float* out = (float*)d_out;
  const int Nn = in_sizes[1];

  char* ws = (char*)d_ws;
  size_t off = 0;
  auto alloc = [&](size_t bytes) -> void* {
    void* p = ws + off;
    off += (bytes + 255) & ~(size_t)255;
    return p;
  };
  float* Apad = (float*)alloc((size_t)G_GRAPHS * KP * sizeof(float));      // 30.4 MB
  float* Wp   = (float*)alloc((size_t)KP * D_LAT * sizeof(float));         //  3.3 MB
  float* H0   = (float*)alloc((size_t)G_GRAPHS * D_LAT * sizeof(float));   //  7.3 MB
  float* A2   = (float*)alloc((size_t)G_GRAPHS * D_LAT * sizeof(float));   //  7.3 MB
  float* H2   = (float*)alloc((size_t)G_GRAPHS * 128 * sizeof(float));     //  2.1 MB

  // 1) pool shell-0 nodes per graph, build features, LayerNorm1
  pool_ln_kernel<<<G_GRAPHS, 448, 0, stream>>>(x, batch, shell, side,
                                               ln1_g, ln1_b, Apad, Nn);
  // 2) pad W_proj to K=1856
  const int totW = KP * D_LAT;
  pad_w_kernel<<<(totW + 255) / 256, 256, 0, stream>>>(W_proj, Wp);
  // 3) GEMM1: H0 = relu(Apad @ Wp + b_proj), M=4096 N=448 K=1856
  dim3 grid1(D_LAT / 64, G_GRAPHS / 32);
  gemm_wmma<true, D_LAT, KP><<<grid1, 128, 0, stream>>>(Apad, Wp, b_proj, H0);
  // 4) LayerNorm2
  ln2_kernel<<<G_GRAPHS, 128, 0, stream>>>(H0, ln2_g, ln2_b, A2);
  // 5) GEMM2: H2 = relu(A2 @ W1 + b1), M=4096 N=128 K=448
  dim3 grid2(128 / 64, G_GRAPHS / 32);
  gemm_wmma<true, 128, D_LAT><<<grid2, 128, 0, stream>>>(A2, W1, b1, H2);
  // 6) final dot with W2 + b2
  readout_kernel<<<G_GRAPHS, 128, 0, stream>>>(H2, W2, b2, out);
}